// DataMakerTorch_26010321944623
// MI455X (gfx1250) — compile-verified
//
#include <hip/hip_runtime.h>
#include <math.h>

#define BATCH 16
#define CHN   3
#define IMH   336
#define IMW   1078
#define HWPIX (IMH * IMW)

typedef __attribute__((ext_vector_type(2))) float v2f;
typedef __attribute__((ext_vector_type(8))) float v8f;

// ---------------------------------------------------------------------------
// Kernel 1: per-batch DLT homography solve (8x8 Gauss-Jordan, partial pivot).
// One lane per batch; 16 batches total -> trivial cost. Writes H (16 x 9 f32)
// into workspace.
// ---------------------------------------------------------------------------
__global__ void dlt_homo_kernel(const float* __restrict__ src_pt,
                                const float* __restrict__ dst_pt,
                                float* __restrict__ Hws)
{
    int b = threadIdx.x;
    if (b >= BATCH) return;

    const float* sp = src_pt + b * 8;   // 4 points x (x,y)
    const float* dp = dst_pt + b * 8;

    float A[8][9];                      // augmented [A | M3]
    for (int n = 0; n < 4; ++n) {
        float x = sp[2 * n], y = sp[2 * n + 1];
        float u = dp[2 * n], v = dp[2 * n + 1];
        float* r0 = A[2 * n];
        float* r1 = A[2 * n + 1];
        r0[0] = x;   r0[1] = y;   r0[2] = 1.f;
        r0[3] = 0.f; r0[4] = 0.f; r0[5] = 0.f;
        r0[6] = -u * x; r0[7] = -u * y; r0[8] = u;
        r1[0] = 0.f; r1[1] = 0.f; r1[2] = 0.f;
        r1[3] = x;   r1[4] = y;   r1[5] = 1.f;
        r1[6] = -v * x; r1[7] = -v * y; r1[8] = v;
    }

    // Gauss-Jordan with partial pivoting (matches inv(A)@M3 numerically well).
    for (int k = 0; k < 8; ++k) {
        int piv = k;
        float best = fabsf(A[k][k]);
        for (int r = k + 1; r < 8; ++r) {
            float m = fabsf(A[r][k]);
            if (m > best) { best = m; piv = r; }
        }
        if (piv != k) {
            for (int c = k; c < 9; ++c) {
                float t = A[k][c]; A[k][c] = A[piv][c]; A[piv][c] = t;
            }
        }
        float inv = 1.0f / A[k][k];
        for (int c = k; c < 9; ++c) A[k][c] *= inv;
        for (int r = 0; r < 8; ++r) {
            if (r == k) continue;
            float f = A[r][k];
            for (int c = k; c < 9; ++c) A[r][c] -= f * A[k][c];
        }
    }

    float* h = Hws + b * 9;
    for (int i = 0; i < 8; ++i) h[i] = A[i][8];
    h[8] = 1.0f;
}

// ---------------------------------------------------------------------------
// Kernel 2: homography warp + bilinear sample.
//
// Coordinate transform (the reference's einsum('bij,jhw->bihw')) is done with
// V_WMMA_F32_16X16X4_F32:
//   A (16x4) = H rows (M=0..2 valid, rest zero)
//   B (4x16) = columns [x; y; 1; 0] for 16 pixels
//   D (16x16): row m, col n = warped coord m of pixel n.
// ISA layout: D VGPR m, lanes 0-15 holds row m (N = lane). So lane n gets
// x'(=d[0]), y'(=d[1]), T(=d[2]) of pixel n in-lane. Two WMMAs + shuffles
// give 32 pixels/wave, one per lane. EXEC is all-1s through the WMMAs
// (tail is handled by clamped index math + predicated loads/stores after).
// ---------------------------------------------------------------------------
__global__ void __launch_bounds__(256)
warp_homography_kernel(const float* __restrict__ src,
                       const float* __restrict__ Hws,
                       float* __restrict__ out)
{
    const int b        = blockIdx.y;
    const int lane     = threadIdx.x & 31;
    const int n        = lane & 15;               // column within a 16-pixel tile
    const bool lo      = lane < 16;
    const int waveBase = blockIdx.x * blockDim.x + (threadIdx.x & ~31);

    const float* __restrict__ Hb = Hws + b * 9;

    // --- A fragment: 16x4 f32. VGPR0 = {K=0 | K=2}, VGPR1 = {K=1 | K=3}.
    // lanes 0-15 (M=n):  a.x = H[n][0], a.y = H[n][1]
    // lanes 16-31 (M=n): a.x = H[n][2], a.y = 0     (K=3 column is zero)
    v2f afrag;
    afrag.x = (n < 3) ? Hb[n * 3 + (lo ? 0 : 2)] : 0.0f;
    afrag.y = (lo && n < 3) ? Hb[n * 3 + 1] : 0.0f;

    // --- B fragments: 4x16 f32. VGPR0 = {K=0 row | K=2 row}, VGPR1 = {K=1 | K=3}.
    // K=0 -> x, K=1 -> y, K=2 -> 1, K=3 -> 0.
    int p0 = waveBase + n;          // pixels 0..15 of this wave
    int p1 = waveBase + 16 + n;     // pixels 16..31 of this wave
    int pc0 = p0 < (HWPIX - 1) ? p0 : (HWPIX - 1);
    int pc1 = p1 < (HWPIX - 1) ? p1 : (HWPIX - 1);
    float fx0 = (float)(pc0 % IMW), fy0 = (float)(pc0 / IMW);
    float fx1 = (float)(pc1 % IMW), fy1 = (float)(pc1 / IMW);

    v2f bfrag0, bfrag1;
    bfrag0.x = lo ? fx0 : 1.0f;   bfrag0.y = lo ? fy0 : 0.0f;
    bfrag1.x = lo ? fx1 : 1.0f;   bfrag1.y = lo ? fy1 : 0.0f;

    v8f cz = {};
    // D = A x B + 0  (full f32 precision -- required: coords up to 1078 need
    // sub-pixel accuracy, so f16/bf16 WMMA would be numerically wrong)
    v8f d0 = __builtin_amdgcn_wmma_f32_16x16x4_f32(
        false, afrag, false, bfrag0, (short)0, cz, false, false);
    v8f d1 = __builtin_amdgcn_wmma_f32_16x16x4_f32(
        false, afrag, false, bfrag1, (short)0, cz, false, false);

    // Cross the second tile's results into lanes 16-31 (uniform shuffles).
    float x1s = __shfl(d1[0], n, 32);
    float y1s = __shfl(d1[1], n, 32);
    float t1s = __shfl(d1[2], n, 32);

    float xw = lo ? d0[0] : x1s;
    float yw = lo ? d0[1] : y1s;
    float tw = lo ? d0[2] : t1s;

    // --- bilinear sample (exact reference formulas) ---
    const int p = waveBase + lane;               // this lane's pixel
    const bool valid = p < HWPIX;

    float t = tw + ((fabsf(tw) >= 1e-7f) ? 0.0f : 1e-6f);
    float xr = xw / t;
    float yr = yw / t;

    float xf = floorf(xr), yf = floorf(yr);
    float x0f = fminf(fmaxf(xf,        0.0f), (float)(IMW - 1));
    float x1f = fminf(fmaxf(xf + 1.0f, 0.0f), (float)(IMW - 1));
    float y0f = fminf(fmaxf(yf,        0.0f), (float)(IMH - 1));
    float y1f = fminf(fmaxf(yf + 1.0f, 0.0f), (float)(IMH - 1));

    float wa = (x1f - xr) * (y1f - yr);
    float wb = (x1f - xr) * (yr - y0f);
    float wc = (xr - x0f) * (y1f - yr);
    float wd = (xr - x0f) * (yr - y0f);

    int ix0 = (int)x0f, ix1 = (int)x1f;
    int iy0 = (int)y0f, iy1 = (int)y1f;

    if (valid) {
        const float* __restrict__ sb = src + (size_t)b * CHN * HWPIX;
        float* __restrict__ ob = out + (size_t)b * CHN * HWPIX + p;
        int o00 = iy0 * IMW + ix0;   // Ia: (y0, x0) -> wa
        int o10 = iy1 * IMW + ix0;   // Ib: (y1, x0) -> wb
        int o01 = iy0 * IMW + ix1;   // Ic: (y0, x1) -> wc
        int o11 = iy1 * IMW + ix1;   // Id: (y1, x1) -> wd
#pragma unroll
        for (int c = 0; c < CHN; ++c) {
            const float* __restrict__ pl = sb + c * HWPIX;
            float r = wa * pl[o00] + wb * pl[o10] + wc * pl[o01] + wd * pl[o11];
            ob[c * HWPIX] = r;
        }
    }
}

// ---------------------------------------------------------------------------
extern "C" void kernel_launch(void* const* d_in, const int* in_sizes, int n_in,
                              void* d_out, int out_size, void* d_ws, size_t ws_size,
                              hipStream_t stream)
{
    const float* src    = (const float*)d_in[0];   // (B, C, H, W) f32
    const float* src_pt = (const float*)d_in[1];   // (B, 4, 2)   f32
    const float* dst_pt = (const float*)d_in[2];   // (B, 4, 2)   f32
    float*       out    = (float*)d_out;           // (B, C, H, W) f32
    float*       Hws    = (float*)d_ws;            // 16 x 9 f32 scratch

    dlt_homo_kernel<<<1, 32, 0, stream>>>(src_pt, dst_pt, Hws);

    dim3 grid((HWPIX + 255) / 256, BATCH);
    warp_homography_kernel<<<grid, 256, 0, stream>>>(src, Hws, out);
}